// interout_group_attention_6030134083982
// MI455X (gfx1250) — compile-verified
//
#include <hip/hip_runtime.h>

// ---------------------------------------------------------------------------
// interout_group_attention for MI455X (gfx1250, wave32, WMMA)
//
// Algebraic simplification: _lw() = sum(softmax(x,-1),-1) == 1 exactly, so
// weightu_all and weightu_behavior are all-ones; only the "weights" FC branch
// (gather-mean -> fc1 GEMM [8x4096]x[4096x2048] -> relu -> fc2 dot) is live.
// That branch is memory-bound (~52 MB total -> ~2.2us @ 23.3 TB/s); the GEMM
// runs on the fp32 WMMA pipe (v_wmma_f32_16x16x4_f32) with an 8-way K-split
// for HBM-saturating concurrency and deterministic partial reduction.
// ---------------------------------------------------------------------------

typedef __attribute__((ext_vector_type(2))) float v2f;
typedef __attribute__((ext_vector_type(8))) float v8f;

#define B_N    3
#define GS_N   8
#define U_N    100000
#define G_N    4096
#define H_N    128
#define M_N    80000
#define N_FC   2048            // G/2
#define MPAD   16              // GS padded to WMMA tile height
#define KSPLIT 8
#define KCHUNK (G_N / KSPLIT)  // 512

#define OUT_ONES  (B_N * GS_N * G_N + B_N * M_N)  // 338304 ones
#define OUT_W_OFF OUT_ONES                        // 24 weights after that

// ws layout (floats): A[16][4096] @ 0 ; partials[128][8][256] @ 65536
#define WS_A_ELEMS   (MPAD * G_N)          // 65536
#define WS_PART_OFF  WS_A_ELEMS

// ---------------------------------------------------------------------------
__global__ void fill_ones_kernel(float* __restrict__ out, int n) {
  int i = blockIdx.x * blockDim.x + threadIdx.x;
  if (i < n) out[i] = 1.0f;
}

__global__ void zero_ws_kernel(float* __restrict__ p, int n) {
  int i = blockIdx.x * blockDim.x + threadIdx.x;
  if (i < n) p[i] = 0.0f;
}

// tempembed[gs,g] = mean_H( user_embedding_gk[B-1, gs, uid[gs,g], :] )
// One wave per (gs,g) row: 32 lanes x float4 = 128 floats, wave reduce.
__global__ void tempembed_kernel(const float* __restrict__ gk,
                                 const int* __restrict__ uidg,
                                 float* __restrict__ A) {
  int wave = threadIdx.x >> 5;
  int lane = threadIdx.x & 31;
  int row  = blockIdx.x * 8 + wave;          // 0 .. GS*G-1
  int gs   = row >> 12;                      // row / 4096
  int uid  = uidg[row];
  const float* src =
      gk + (((size_t)((B_N - 1) * GS_N + gs) * (size_t)U_N) + (size_t)uid) * H_N;
  float4 v = ((const float4*)src)[lane];
  float s = v.x + v.y + v.z + v.w;
#pragma unroll
  for (int m = 16; m > 0; m >>= 1) s += __shfl_xor(s, m, 32);
  if (lane == 0) A[row] = s * (1.0f / 128.0f);
}

// fc1 GEMM tile: D[16,16] = A[16, Kchunk] x W[Kchunk, 16] via v_wmma_f32_16x16x4_f32.
// grid = (128 n-tiles, 8 k-splits), 32 threads (one wave) per block.
__global__ __launch_bounds__(32)
void fc1_wmma_kernel(const float* __restrict__ A,
                     const float* __restrict__ W,
                     float* __restrict__ part) {
  const int lane  = threadIdx.x;
  const int ntile = blockIdx.x;          // 0..127
  const int ks    = blockIdx.y;          // 0..7
  const int n0    = ntile * 16;
  const int m     = lane & 15;           // row for A-frag, col for B-frag
  const int kh    = (lane >> 4) * 2;     // lane-half K offset: 0 or 2

  const float* Arow = A + (size_t)m * G_N;
  const int k0 = ks * KCHUNK;

  v8f c = {};
#pragma unroll 4
  for (int k = k0; k < k0 + KCHUNK; k += 4) {
    // A 16x4 f32 frag: lanes 0-15 hold K={k,k+1}, lanes 16-31 hold K={k+2,k+3}
    v2f a;
    a[0] = Arow[k + kh];
    a[1] = Arow[k + kh + 1];
    // B 4x16 f32 frag: lane col = m, lane-half rows kh / kh+1
    const float* Wr = W + (size_t)(k + kh) * N_FC + n0 + m;
    v2f b;
    b[0] = Wr[0];
    b[1] = Wr[N_FC];
    // prefetch the B stream ~16 K-rows ahead (speculative, clamped)
    int kpf = k + 16; kpf = kpf < (G_N - 1) ? kpf : (G_N - 1);
    __builtin_prefetch(W + (size_t)kpf * N_FC + n0, 0, 0);

    c = __builtin_amdgcn_wmma_f32_16x16x4_f32(
        /*neg_a=*/false, a, /*neg_b=*/false, b,
        /*c_mod=*/(short)0, c, /*reuse_a=*/false, /*reuse_b=*/false);
  }

  // store partial tile; D layout: reg r, lanes 0-15 -> D[r, lane] (rows 0-7 = gs)
  float* dst = part + (size_t)(ntile * KSPLIT + ks) * 256 + lane;
#pragma unroll
  for (int r = 0; r < 8; ++r) dst[r * 32] = c[r];
}

// w[gs] = sum_n relu(sum_ks partial + fc1_b[n]) * fc2_W[n] + fc2_b
// Deterministic: fixed K-split summation order, LDS tree reduction.
__global__ void finalize_kernel(const float* __restrict__ part,
                                const float* __restrict__ fc1b,
                                const float* __restrict__ fc2W,
                                const float* __restrict__ fc2b,
                                float* __restrict__ out) {
  __shared__ float red[256];
  const int gs = blockIdx.x;   // 0..7
  const int t  = threadIdx.x;  // 0..255
  float acc = 0.0f;
  for (int n = t; n < N_FC; n += 256) {
    const int ntile = n >> 4;
    const int nn    = n & 15;
    float s = 0.0f;
#pragma unroll
    for (int ksp = 0; ksp < KSPLIT; ++ksp)
      s += part[(size_t)(ntile * KSPLIT + ksp) * 256 + gs * 32 + nn];
    s += fc1b[n];
    s = s > 0.0f ? s : 0.0f;   // relu
    acc += s * fc2W[n];
  }
  red[t] = acc;
  __syncthreads();
  for (int off = 128; off > 0; off >>= 1) {
    if (t < off) red[t] += red[t + off];
    __syncthreads();
  }
  if (t == 0) {
    float w = red[0] + fc2b[0];
#pragma unroll
    for (int b = 0; b < B_N; ++b) out[OUT_W_OFF + b * GS_N + gs] = w;
  }
}

// ---------------------------------------------------------------------------
extern "C" void kernel_launch(void* const* d_in, const int* in_sizes, int n_in,
                              void* d_out, int out_size, void* d_ws, size_t ws_size,
                              hipStream_t stream) {
  (void)in_sizes; (void)n_in; (void)out_size; (void)ws_size;

  const float* gk   = (const float*)d_in[1];   // user_embedding_gk [B,GS,U,H]
  const int*   uidg = (const int*)  d_in[4];   // user_id_groups [GS,G]
  const float* fc1W = (const float*)d_in[30];  // [4096,2048]
  const float* fc1b = (const float*)d_in[31];  // [2048]
  const float* fc2W = (const float*)d_in[32];  // [2048,1]
  const float* fc2b = (const float*)d_in[33];  // [1]

  float* out  = (float*)d_out;
  float* A    = (float*)d_ws;            // [16][4096] tempembed (rows 8-15 zero)
  float* part = A + WS_PART_OFF;         // [128][8][256] K-split partials

  // 1) ones for weightu_all + weightu_behavior (algebraically exact)
  fill_ones_kernel<<<(OUT_ONES + 255) / 256, 256, 0, stream>>>(out, OUT_ONES);
  // 2) zero A pad rows 8..15
  zero_ws_kernel<<<(WS_A_ELEMS / 2 + 255) / 256, 256, 0, stream>>>(
      A + GS_N * G_N, WS_A_ELEMS / 2);
  // 3) gather + mean over H  -> A rows 0..7
  tempembed_kernel<<<(GS_N * G_N) / 8, 256, 0, stream>>>(gk, uidg, A);
  // 4) fc1 GEMM on the f32 WMMA pipe, 128 n-tiles x 8 k-splits
  fc1_wmma_kernel<<<dim3(N_FC / 16, KSPLIT), 32, 0, stream>>>(A, fc1W, part);
  // 5) deterministic K-reduction + bias + relu + fc2 + broadcast to [B,GS,1]
  finalize_kernel<<<GS_N, 256, 0, stream>>>(part, fc1b, fc2W, fc2b, out);
}